// DehazingLayer_43327630082141
// MI455X (gfx1250) — compile-verified
//
#include <hip/hip_runtime.h>

typedef float v2f __attribute__((ext_vector_type(2)));
typedef float v4f __attribute__((ext_vector_type(4)));
typedef float v8f __attribute__((ext_vector_type(8)));

#define B_   64
#define H_   256
#define W_   256
#define NPIX (B_ * H_ * W_)
#define GFR  40            // guided filter radius
#define GFEPS 1e-3f

__device__ __forceinline__ float cntwin(int i) {
    int lo = i - GFR; if (lo < 0) lo = 0;
    int hi = i + GFR; if (hi > 255) hi = 255;
    return (float)(hi - lo + 1);
}

// ---------------- elementwise prologue ----------------
__global__ void k_initA(unsigned* Abits) {
    int t = threadIdx.x;
    if (t < B_ * 3) Abits[t] = 0u;
}

__global__ void k_minc_luma(const float* __restrict__ x, float* __restrict__ minc,
                            float* __restrict__ g) {
    int idx = blockIdx.x * blockDim.x + threadIdx.x;
    float x0 = x[idx * 3 + 0], x1 = x[idx * 3 + 1], x2 = x[idx * 3 + 2];
    minc[idx] = fminf(x0, fminf(x1, x2));
    g[idx] = 0.299f * x0 + 0.587f * x1 + 0.114f * x2;
}

// ---------------- separable clipped min filter ----------------
__global__ void k_rowmin(const float* __restrict__ in, float* __restrict__ out,
                         const int* __restrict__ radius) {
    int idx = blockIdx.x * blockDim.x + threadIdx.x;
    int w = idx & (W_ - 1);
    int base = idx - w;
    int win = radius[0];
    int lo_off = (win - 1) >> 1, hi_off = win - 1 - lo_off;   // SAME padding
    int lo = w - lo_off; if (lo < 0) lo = 0;
    int hi = w + hi_off; if (hi > W_ - 1) hi = W_ - 1;
    float m = 3.4e38f;
    for (int ww = lo; ww <= hi; ++ww) m = fminf(m, in[base + ww]);
    out[idx] = m;
}

__global__ void k_colmin(const float* __restrict__ in, float* __restrict__ out,
                         const int* __restrict__ radius, int affine,
                         const float* __restrict__ omega) {
    int idx = blockIdx.x * blockDim.x + threadIdx.x;
    int w = idx & (W_ - 1);
    int h = (idx >> 8) & (H_ - 1);
    int b = idx >> 16;
    int win = radius[0];
    int lo_off = (win - 1) >> 1, hi_off = win - 1 - lo_off;
    int lo = h - lo_off; if (lo < 0) lo = 0;
    int hi = h + hi_off; if (hi > H_ - 1) hi = H_ - 1;
    float m = 3.4e38f;
    for (int hh = lo; hh <= hi; ++hh) m = fminf(m, in[((b << 8) + hh) * W_ + w]);
    if (affine) m = 1.0f - omega[0] * m;   // t_est = 1 - omega * dark_n
    out[idx] = m;
}

// ---------------- exact 0.999-quantile via MSB radix-select ----------------
// One block per batch. dark >= 0 so float bit patterns are order-preserving.
__global__ void k_select_thr(const float* __restrict__ dark, float* __restrict__ thr) {
    __shared__ unsigned hist[256];
    __shared__ unsigned s_prefix, s_k;
    const int n = H_ * W_;
    const float* db = dark + blockIdx.x * n;
    int tid = threadIdx.x;

    float idxf = 0.999f * (float)(n - 1);
    float flo  = floorf(idxf);
    float frac = idxf - flo;
    unsigned rank_lo = (unsigned)n - (unsigned)flo;   // value at ascending index floor
    unsigned rank_hi = rank_lo - 1u;                  // ascending index floor+1 (larger value)

    float v[2];
    for (int t = 0; t < 2; ++t) {
        unsigned k = (t == 0) ? rank_hi : rank_lo;    // k-th largest
        unsigned prefix = 0u;
        for (int shift = 24; shift >= 0; shift -= 8) {
            hist[tid] = 0u;
            __syncthreads();
            unsigned himask = (shift == 24) ? 0u : (0xFFFFFFFFu << (shift + 8));
            for (int i = tid; i < n; i += 256) {
                unsigned key = __float_as_uint(db[i]);
                if ((key & himask) == prefix)
                    atomicAdd(&hist[(key >> shift) & 255u], 1u);
            }
            __syncthreads();
            if (tid == 0) {
                unsigned cum = 0;
                for (int bb = 255; bb >= 0; --bb) {
                    cum += hist[bb];
                    if (cum >= k) {
                        s_prefix = prefix | ((unsigned)bb << shift);
                        s_k = k - (cum - hist[bb]);
                        break;
                    }
                }
            }
            __syncthreads();
            prefix = s_prefix;
            k = s_k;
            __syncthreads();
        }
        v[t] = __uint_as_float(prefix);
    }
    if (tid == 0) thr[blockIdx.x] = v[1] + frac * (v[0] - v[1]);
}

// ---------------- atmospheric light ----------------
__global__ void k_amax(const float* __restrict__ x, const float* __restrict__ dark,
                       const float* __restrict__ thr, unsigned* __restrict__ Abits) {
    int idx = blockIdx.x * blockDim.x + threadIdx.x;
    int b = idx >> 16;
    if (dark[idx] >= thr[b]) {
        atomicMax(&Abits[b * 3 + 0], __float_as_uint(x[idx * 3 + 0]));
        atomicMax(&Abits[b * 3 + 1], __float_as_uint(x[idx * 3 + 1]));
        atomicMax(&Abits[b * 3 + 2], __float_as_uint(x[idx * 3 + 2]));
    }
}

__global__ void k_finalizeA(unsigned* Abits) {
    int t = threadIdx.x;
    if (t < B_ * 3) {
        float v = __uint_as_float(Abits[t]);
        ((float*)Abits)[t] = fmaxf(v, 1e-3f);
    }
}

__global__ void k_mincn(const float* __restrict__ x, const float* __restrict__ A,
                        float* __restrict__ out) {
    int idx = blockIdx.x * blockDim.x + threadIdx.x;
    const float* Ab = A + (idx >> 16) * 3;
    float m = fminf(x[idx * 3 + 0] / Ab[0],
               fminf(x[idx * 3 + 1] / Ab[1], x[idx * 3 + 2] / Ab[2]));
    out[idx] = m;
}

// ------- banded row-box GEMM with transposed store (WMMA f32 16x16x4) -------
// Computes Dt = (S * K)^T per batch image, K[i,j] = 1{|i-j|<=GFR} generated
// on the fly on the B operand (zero memory traffic).  Applying this kernel
// twice yields (K * S * K) = full 2D clipped box sum (K symmetric):
//   pass1: Y^T = (S*K)^T ;  pass2: (Y^T*K)^T = K*S*K.
// Each wave owns a 64x16 output block: 4 M-tiles sharing one generated band
// operand per k-step -> 4 clustered b64 loads (immediate offsets) + 4
// back-to-back WMMAs per iteration, indicator/loop overhead amortized 4x.
// SRCMUL=1 fuses the elementwise product s0*s1 on the A operand (g*p, g*g).
template <int SRCMUL>
__global__ void k_box_gemm_t(const float* __restrict__ s0, const float* __restrict__ s1,
                             float* __restrict__ dst) {
    const int RSTRIDE = 16 << 8;       // 16 rows = 16*256 floats
    int lane = threadIdx.x & 31;
    int wave = threadIdx.x >> 5;
    int gt = blockIdx.x * 8 + wave;    // 64 wave-jobs per image (4 Mblk x 16 Ntile)
    int b  = gt >> 6;
    int t  = gt & 63;
    int h0 = (t >> 4) << 6;            // M-block base (64 rows)
    int w0 = (t & 15) << 4;            // N-tile base = band center
    int base = b << 16;

    int m  = lane & 15;                // A row slot == B col slot (N)
    int kh = lane >> 4;                // K-half: this lane holds K = k0+2*kh+{0,1}
    int koff = 2 * kh;

    int klo = w0 - GFR; if (klo < 0) klo = 0;
    klo &= ~3;                          // align to WMMA K-step
    int khi = w0 + 16 + GFR;            // exclusive; multiple of 4 after clamp
    if (khi > 256) khi = 256;

    const float* r0 = s0 + base + ((h0 + m) << 8) + koff;
    const float* r1 = s1 + base + ((h0 + m) << 8) + koff;

    int d0 = klo + koff - (w0 + m) + GFR;   // in-band iff (unsigned)d <= 2*GFR
    v8f acc0 = {}, acc1 = {}, acc2 = {}, acc3 = {};
    for (int k0 = klo; k0 < khi; k0 += 4) {
        v2f a0 = *(const v2f*)(r0 + k0);
        v2f a1 = *(const v2f*)(r0 + k0 + RSTRIDE);
        v2f a2 = *(const v2f*)(r0 + k0 + 2 * RSTRIDE);
        v2f a3 = *(const v2f*)(r0 + k0 + 3 * RSTRIDE);
        if (SRCMUL) {
            a0 *= *(const v2f*)(r1 + k0);
            a1 *= *(const v2f*)(r1 + k0 + RSTRIDE);
            a2 *= *(const v2f*)(r1 + k0 + 2 * RSTRIDE);
            a3 *= *(const v2f*)(r1 + k0 + 3 * RSTRIDE);
        }
        v2f bv;
        bv.x = ((unsigned)d0 <= 2u * GFR) ? 1.0f : 0.0f;
        bv.y = ((unsigned)(d0 + 1) <= 2u * GFR) ? 1.0f : 0.0f;
        acc0 = __builtin_amdgcn_wmma_f32_16x16x4_f32(false, a0, false, bv, (short)0, acc0, false, false);
        acc1 = __builtin_amdgcn_wmma_f32_16x16x4_f32(false, a1, false, bv, (short)0, acc1, false, false);
        acc2 = __builtin_amdgcn_wmma_f32_16x16x4_f32(false, a2, false, bv, (short)0, acc2, false, false);
        acc3 = __builtin_amdgcn_wmma_f32_16x16x4_f32(false, a3, false, bv, (short)0, acc3, false, false);
        d0 += 4;
    }

    // transposed store: Dt[N][M]; accJ[i] = (M = h0+16*J+8*kh+i, N = w0+m)
    // -> lanes m, m+16 together own 64 consecutive floats of one column: 8 b128 stores per lane
    float* Dt = dst + base + ((w0 + m) << 8) + h0 + (kh << 3);
    v4f q;
    q = (v4f){acc0[0], acc0[1], acc0[2], acc0[3]}; *(v4f*)(Dt +  0) = q;
    q = (v4f){acc0[4], acc0[5], acc0[6], acc0[7]}; *(v4f*)(Dt +  4) = q;
    q = (v4f){acc1[0], acc1[1], acc1[2], acc1[3]}; *(v4f*)(Dt + 16) = q;
    q = (v4f){acc1[4], acc1[5], acc1[6], acc1[7]}; *(v4f*)(Dt + 20) = q;
    q = (v4f){acc2[0], acc2[1], acc2[2], acc2[3]}; *(v4f*)(Dt + 32) = q;
    q = (v4f){acc2[4], acc2[5], acc2[6], acc2[7]}; *(v4f*)(Dt + 36) = q;
    q = (v4f){acc3[0], acc3[1], acc3[2], acc3[3]}; *(v4f*)(Dt + 48) = q;
    q = (v4f){acc3[4], acc3[5], acc3[6], acc3[7]}; *(v4f*)(Dt + 52) = q;
}

// ---------------- guided filter coefficients ----------------
__global__ void k_ab(const float* __restrict__ bg, const float* __restrict__ bp,
                     float* __restrict__ bgp, float* __restrict__ bgg) {
    int idx = blockIdx.x * blockDim.x + threadIdx.x;
    int w = idx & 255, h = (idx >> 8) & 255;
    float Ninv = 1.0f / (cntwin(h) * cntwin(w));
    float mg = bg[idx] * Ninv, mp = bp[idx] * Ninv;
    float cov = bgp[idx] * Ninv - mg * mp;
    float var = bgg[idx] * Ninv - mg * mg;
    float a = cov / (var + GFEPS);
    float bb = mp - a * mg;
    bgp[idx] = a;
    bgg[idx] = bb;
}

// ---------------- radiance recovery ----------------
__global__ void k_final(const float* __restrict__ ba, const float* __restrict__ bb,
                        const float* __restrict__ g, const float* __restrict__ x,
                        const float* __restrict__ A, const float* __restrict__ tminp,
                        float* __restrict__ out) {
    int idx = blockIdx.x * blockDim.x + threadIdx.x;
    int w = idx & 255, h = (idx >> 8) & 255, b = idx >> 16;
    float Ninv = 1.0f / (cntwin(h) * cntwin(w));
    float q = ba[idx] * Ninv * g[idx] + bb[idx] * Ninv;
    float t = fmaxf(q, tminp[0]);
    const float* Ab = A + b * 3;
    #pragma unroll
    for (int c = 0; c < 3; ++c) {
        float Ac = Ab[c];
        float J = (x[idx * 3 + c] - Ac) / t + Ac;
        out[idx * 3 + c] = fminf(fmaxf(J, 0.0f), 1.0f);
    }
}

extern "C" void kernel_launch(void* const* d_in, const int* in_sizes, int n_in,
                              void* d_out, int out_size, void* d_ws, size_t ws_size,
                              hipStream_t stream) {
    (void)in_sizes; (void)n_in; (void)out_size; (void)ws_size;
    const float* x      = (const float*)d_in[0];
    const float* tmin   = (const float*)d_in[1];
    const float* omega  = (const float*)d_in[2];
    const int*   radius = (const int*)d_in[3];
    float* out = (float*)d_out;

    // ws: 4 planes + thr + A  (~67 MB)
    float* G   = (float*)d_ws;
    float* P   = G + NPIX;
    float* B1  = P + NPIX;
    float* B2  = B1 + NPIX;
    float* thr = B2 + NPIX;                    // 64 floats
    unsigned* Abits = (unsigned*)(thr + 64);   // 192 entries

    // d_out used as scratch until the final kernel rewrites it
    float* T  = out;             // GEMM temp / channel-min temp
    float* O1 = out + NPIX;      // box(g*p) -> a
    float* O2 = out + 2 * NPIX;  // box(g*g) -> b

    dim3 blk(256);
    dim3 grd(NPIX / 256);
    dim3 ggrd(B_ * 64 / 8);      // 8 wave-jobs (64x16 blocks) per 256-thread block

    k_initA<<<1, 256, 0, stream>>>(Abits);
    k_minc_luma<<<grd, blk, 0, stream>>>(x, T, G);
    k_rowmin<<<grd, blk, 0, stream>>>(T, B1, radius);
    k_colmin<<<grd, blk, 0, stream>>>(B1, B2, radius, 0, omega);     // dark
    k_select_thr<<<B_, 256, 0, stream>>>(B2, thr);
    k_amax<<<grd, blk, 0, stream>>>(x, B2, thr, Abits);
    k_finalizeA<<<1, 256, 0, stream>>>(Abits);
    k_mincn<<<grd, blk, 0, stream>>>(x, (const float*)Abits, T);
    k_rowmin<<<grd, blk, 0, stream>>>(T, B1, radius);
    k_colmin<<<grd, blk, 0, stream>>>(B1, P, radius, 1, omega);      // p = t_est

    // Box(X) = two applications of the transposed-store row-box GEMM
    // Box(g) -> B1
    k_box_gemm_t<0><<<ggrd, blk, 0, stream>>>(G, G, T);
    k_box_gemm_t<0><<<ggrd, blk, 0, stream>>>(T, T, B1);
    // Box(p) -> B2
    k_box_gemm_t<0><<<ggrd, blk, 0, stream>>>(P, P, T);
    k_box_gemm_t<0><<<ggrd, blk, 0, stream>>>(T, T, B2);
    // Box(g*p) -> O1 (product fused on A operand)
    k_box_gemm_t<1><<<ggrd, blk, 0, stream>>>(G, P, T);
    k_box_gemm_t<0><<<ggrd, blk, 0, stream>>>(T, T, O1);
    // Box(g*g) -> O2
    k_box_gemm_t<1><<<ggrd, blk, 0, stream>>>(G, G, T);
    k_box_gemm_t<0><<<ggrd, blk, 0, stream>>>(T, T, O2);

    k_ab<<<grd, blk, 0, stream>>>(B1, B2, O1, O2);

    // Box(a) -> B1 ; Box(b) -> B2
    k_box_gemm_t<0><<<ggrd, blk, 0, stream>>>(O1, O1, T);
    k_box_gemm_t<0><<<ggrd, blk, 0, stream>>>(T, T, B1);
    k_box_gemm_t<0><<<ggrd, blk, 0, stream>>>(O2, O2, T);
    k_box_gemm_t<0><<<ggrd, blk, 0, stream>>>(T, T, B2);

    k_final<<<grd, blk, 0, stream>>>(B1, B2, G, x, (const float*)Abits, tmin, out);
}